// ActorCriticNetwork_85160611545655
// MI455X (gfx1250) — compile-verified
//
#include <hip/hip_runtime.h>
#include <math.h>

// ---------------- problem constants ----------------
#define B_    8192      // batch
#define F_    128       // features
#define N_    101       // nodes
#define NQ    303       // n = 3N
#define NPAD  320       // n padded to multiple of 32 (2x2 WMMA tile pairs)
#define KLOOP 304       // K extent actually needed (rhs cols >= 303 are zero)
#define MROWS 505       // m = m_eq + n
#define MPAD  512       // m padded (stride)
#define MEQ_  202       // 2*(N-1)+2
#define GJLD  640       // Gauss-Jordan row stride (2*NPAD)
#define KACC  104       // padded K for acc GEMM (covers x cols 200..303)

constexpr float DT_  = 1.0f / 100.0f;
constexpr float RHO_ = 1.0f;
constexpr float SIG_ = 1e-6f;
constexpr float ALP_ = 1.6f;

typedef __attribute__((ext_vector_type(2))) float v2f;
typedef __attribute__((ext_vector_type(8))) float v8f;

static __device__ __forceinline__ v8f wmma4(v2f a, v2f b, v8f c) {
  // V_WMMA_F32_16X16X4_F32: (neg_a, A, neg_b, B, c_mod, C, reuse_a, reuse_b)
  return __builtin_amdgcn_wmma_f32_16x16x4_f32(false, a, false, b, (short)0, c,
                                               false, false);
}

static __device__ __forceinline__ float softplusf(float v) {
  return v > 20.0f ? v : log1pf(expf(v));
}

// ---------------- generic zero fill ----------------
__global__ void zero_kernel(float* __restrict__ p, size_t n) {
  size_t i = (size_t)blockIdx.x * blockDim.x + threadIdx.x;
  size_t stride = (size_t)gridDim.x * blockDim.x;
  for (; i < n; i += stride) p[i] = 0.0f;
}

// ---------------- build A (505 x NPAD, pre-zeroed) ----------------
__global__ void build_A_kernel(float* __restrict__ Amat) {
  int r = blockIdx.x * blockDim.x + threadIdx.x;
  if (r >= MROWS) return;
  float* row = Amat + (size_t)r * NPAD;
  const float hdt = 0.5f * DT_;
  if (r < N_ - 1) {                       // pos dynamics
    int i = r;
    row[i + 1] = 1.0f; row[i] = -1.0f;
    row[N_ + i] = -hdt; row[N_ + i + 1] = -hdt;
  } else if (r < 2 * (N_ - 1)) {          // vel dynamics
    int i = r - (N_ - 1);
    row[N_ + i + 1] = 1.0f; row[N_ + i] = -1.0f;
    row[2 * N_ + i] = -hdt; row[2 * N_ + i + 1] = -hdt;
  } else if (r == MEQ_ - 2) {
    row[0] = 1.0f;
  } else if (r == MEQ_ - 1) {
    row[N_] = 1.0f;
  } else {
    row[r - MEQ_] = 1.0f;                 // identity block
  }
}

// ---------------- build augmented [K | I] ----------------
__global__ void build_K_kernel(const float* __restrict__ Amat, float* __restrict__ gj) {
  int t = blockIdx.x * blockDim.x + threadIdx.x;
  if (t >= NPAD * NPAD) return;
  int i = t / NPAD, j = t % NPAD;
  float acc = 0.0f;
  if (i < NQ && j < NQ) {
    for (int r = 0; r < MROWS; ++r)
      acc += Amat[(size_t)r * NPAD + i] * Amat[(size_t)r * NPAD + j];
    acc *= RHO_;
    if (i == j) {
      float pd = (i < N_) ? 2.0f : ((i >= 2 * N_) ? 0.02f : 0.0f);
      acc += pd + SIG_;
    }
  } else if (i == j) {
    acc = 1.0f;                            // pad diagonal -> Kinv pad col = e_i
  }
  gj[(size_t)i * GJLD + j] = acc;
  gj[(size_t)i * GJLD + NPAD + j] = (i == j) ? 1.0f : 0.0f;
}

// ---------------- single-workgroup Gauss-Jordan (K is SPD) ----------------
__global__ __launch_bounds__(1024) void gauss_jordan_kernel(float* __restrict__ gj) {
  __shared__ float fcol[NPAD];
  const int tid = threadIdx.x;
  for (int p = 0; p < NPAD; ++p) {
    for (int r = tid; r < NPAD; r += 1024) fcol[r] = gj[(size_t)r * GJLD + p];
    __syncthreads();
    const float pinv = 1.0f / fcol[p];
    float* prow = gj + (size_t)p * GJLD;
    for (int c = tid; c < GJLD; c += 1024) prow[c] *= pinv;
    __syncthreads();
    for (int r = 0; r < NPAD; ++r) {
      if (r == p) continue;
      const float f = fcol[r];
      float* row = gj + (size_t)r * GJLD;
      for (int c = tid; c < GJLD; c += 1024) row[c] -= f * prow[c];
    }
    __syncthreads();
  }
}

// ---------------- zero-padded copies of W4/W5 for the acc GEMM ----------------
__global__ void pad_w_kernel(const float* __restrict__ W4, const float* __restrict__ W5,
                             float* __restrict__ W4p, float* __restrict__ W5p) {
  int t = blockIdx.x * blockDim.x + threadIdx.x;
  if (t >= KACC * F_) return;
  int k = t >> 7, f = t & (F_ - 1);
  float a = 0.0f, b = 0.0f;
  if (k >= 2 && k <= 102) {                // x col 200+k: 202..302 are acc[0..100]
    a = W4[(size_t)(k - 2) * F_ + f];
    b = W5[(size_t)(k - 2) * F_ + f];
  }
  W4p[t] = a; W5p[t] = b;
}

// ---------------- first MLP layer (K=2, scalar) ----------------
__global__ void head1_kernel(const float* __restrict__ x0, const float* __restrict__ W1,
                             const float* __restrict__ b1, float* __restrict__ h1) {
  int t = blockIdx.x * blockDim.x + threadIdx.x;
  if (t >= B_ * F_) return;
  int b = t >> 7, f = t & (F_ - 1);
  float v = x0[b * 2 + 0] * W1[f] + x0[b * 2 + 1] * W1[F_ + f] + b1[f];
  h1[t] = tanhf(v);
}

// ------- WMMA GEMM (32x32 tile/wave): C[b][f] = act(A[b][aoff..]*W + bias) -------
__global__ __launch_bounds__(32)
void gemm_feat_kernel(const float* __restrict__ A, int lda, int aoff,
                      const float* __restrict__ W, const float* __restrict__ bias,
                      float* __restrict__ C, int K, int act) {
  const int n0 = blockIdx.x * 32;          // feature tile pair
  const int m0 = blockIdx.y * 32;          // batch tile pair
  const int lane = threadIdx.x;
  const int half = lane >> 4, idx = lane & 15;
  const float* arow0 = A + (size_t)(m0 + idx) * lda + aoff;
  const float* arow1 = arow0 + (size_t)16 * lda;
  const float* bcol0 = W + (n0 + idx);
  const float* bcol1 = bcol0 + 16;
  v8f c00 = {}, c01 = {}, c10 = {}, c11 = {};
  for (int k = 0; k < K; k += 4) {
    const int kk = k + 2 * half;
    v2f a0 = *(const v2f*)(arow0 + kk);
    v2f a1 = *(const v2f*)(arow1 + kk);
    const size_t r0 = (size_t)kk * F_, r1 = r0 + F_;
    v2f b0, b1;
    b0.x = bcol0[r0]; b0.y = bcol0[r1];
    b1.x = bcol1[r0]; b1.y = bcol1[r1];
    c00 = wmma4(a0, b0, c00);
    c01 = wmma4(a0, b1, c01);
    c10 = wmma4(a1, b0, c10);
    c11 = wmma4(a1, b1, c11);
  }
  const float bv0 = bias[n0 + idx];
  const float bv1 = bias[n0 + 16 + idx];
#pragma unroll
  for (int v = 0; v < 8; ++v) {
    const int r0 = m0 + v + 8 * half;
    float t00 = c00[v] + bv0, t01 = c01[v] + bv1;
    float t10 = c10[v] + bv0, t11 = c11[v] + bv1;
    if (act) { t00 = tanhf(t00); t01 = tanhf(t01); t10 = tanhf(t10); t11 = tanhf(t11); }
    C[(size_t)r0 * F_ + n0 + idx]             = t00;
    C[(size_t)r0 * F_ + n0 + 16 + idx]        = t01;
    C[(size_t)(r0 + 16) * F_ + n0 + idx]      = t10;
    C[(size_t)(r0 + 16) * F_ + n0 + 16 + idx] = t11;
  }
}

// ---------------- target = h2 @ Wt + bt ----------------
__global__ __launch_bounds__(128)
void target_kernel(const float* __restrict__ h2, const float* __restrict__ Wt,
                   const float* __restrict__ bt, float* __restrict__ target) {
  __shared__ float red[F_];
  int b = blockIdx.x, t = threadIdx.x;
  red[t] = h2[(size_t)b * F_ + t] * Wt[t];
  __syncthreads();
  for (int s = 64; s > 0; s >>= 1) {
    if (t < s) red[t] += red[t + s];
    __syncthreads();
  }
  if (t == 0) target[b] = red[0] + bt[0];
}

// ---------------- ADMM: rhs = sigma*x - q + A^T (rho*z - y)  (A sparse) --------
__global__ __launch_bounds__(128)
void admm_rhs_kernel(const float* __restrict__ x, const float* __restrict__ z,
                     const float* __restrict__ yv, const float* __restrict__ target,
                     float* __restrict__ rhs) {
  int j = blockIdx.x * blockDim.x + threadIdx.x;
  int b = blockIdx.y;
  if (j >= NPAD) return;
  const size_t xb = (size_t)b * NPAD, zb = (size_t)b * MPAD;
  float out = 0.0f;
  if (j < NQ) {
    const float hdt = 0.5f * DT_;
    auto w = [&](int r) { return RHO_ * z[zb + r] - yv[zb + r]; };
    float acc = SIG_ * x[xb + j] + w(MEQ_ + j);       // identity row
    if (j < N_) {                                     // pos var
      acc += 2.0f * target[b];                        // -q
      if (j >= 1) acc += w(j - 1);
      if (j <= N_ - 2) acc -= w(j);
      if (j == 0) acc += w(2 * N_ - 2);               // initial pos row
    } else if (j < 2 * N_) {                          // vel var
      int v = j - N_;
      if (v >= 1)      { acc -= hdt * w(v - 1); acc += w(N_ - 1 + v - 1); }
      if (v <= N_ - 2) { acc -= hdt * w(v);     acc -= w(N_ - 1 + v); }
      if (v == 0) acc += w(2 * N_ - 1);               // initial vel row
    } else {                                          // acc var
      int a = j - 2 * N_;
      if (a >= 1)      acc -= hdt * w(N_ - 1 + a - 1);
      if (a <= N_ - 2) acc -= hdt * w(N_ - 1 + a);
    }
    out = acc;
  }
  rhs[xb + j] = out;
}

// ------- ADMM: xt = rhs @ KINV (32x32 tile/wave), fused x <- a*xt+(1-a)*x -------
__global__ __launch_bounds__(32)
void admm_gemm_kernel(const float* __restrict__ rhs, const float* __restrict__ gj,
                      float* __restrict__ xt, float* __restrict__ x) {
  const int n0 = blockIdx.x * 32;          // 10 tile-pairs over NPAD
  const int m0 = blockIdx.y * 32;          // 256 tile-pairs over batch
  const int lane = threadIdx.x;
  const int half = lane >> 4, idx = lane & 15;
  const float* arow0 = rhs + (size_t)(m0 + idx) * NPAD;
  const float* arow1 = arow0 + (size_t)16 * NPAD;
  const float* bcol0 = gj + NPAD + (n0 + idx);       // KINV lives in right half
  const float* bcol1 = bcol0 + 16;
  v8f c00 = {}, c01 = {}, c10 = {}, c11 = {};
  for (int k = 0; k < KLOOP; k += 4) {
    const int kk = k + 2 * half;
    if (k + 16 < KLOOP)
      __builtin_prefetch(bcol0 + (size_t)(kk + 16) * GJLD, 0, 3);
    v2f a0 = *(const v2f*)(arow0 + kk);
    v2f a1 = *(const v2f*)(arow1 + kk);
    const size_t r0 = (size_t)kk * GJLD, r1 = r0 + GJLD;
    v2f b0, b1;
    b0.x = bcol0[r0]; b0.y = bcol0[r1];
    b1.x = bcol1[r0]; b1.y = bcol1[r1];
    c00 = wmma4(a0, b0, c00);
    c01 = wmma4(a0, b1, c01);
    c10 = wmma4(a1, b0, c10);
    c11 = wmma4(a1, b1, c11);
  }
#pragma unroll
  for (int v = 0; v < 8; ++v) {
    const int r0 = m0 + v + 8 * half;
    const size_t o00 = (size_t)r0 * NPAD + n0 + idx;
    const size_t o01 = o00 + 16;
    const size_t o10 = o00 + (size_t)16 * NPAD;
    const size_t o11 = o10 + 16;
    float t;
    t = c00[v]; xt[o00] = t; x[o00] = ALP_ * t + (1.0f - ALP_) * x[o00];
    t = c01[v]; xt[o01] = t; x[o01] = ALP_ * t + (1.0f - ALP_) * x[o01];
    t = c10[v]; xt[o10] = t; x[o10] = ALP_ * t + (1.0f - ALP_) * x[o10];
    t = c11[v]; xt[o11] = t; x[o11] = ALP_ * t + (1.0f - ALP_) * x[o11];
  }
}

// ---------------- ADMM: zt = A xt (sparse), project, dual update --------------
__global__ __launch_bounds__(128)
void admm_update_kernel(const float* __restrict__ xt, const float* __restrict__ ic,
                        float* __restrict__ z, float* __restrict__ yv) {
  int r = blockIdx.x * blockDim.x + threadIdx.x;
  int b = blockIdx.y;
  if (r >= MROWS) return;
  const size_t xb = (size_t)b * NPAD, zb = (size_t)b * MPAD;
  const float hdt = 0.5f * DT_;
  float zt, lo, hi;
  if (r < N_ - 1) {
    zt = xt[xb + r + 1] - xt[xb + r] - hdt * (xt[xb + N_ + r] + xt[xb + N_ + r + 1]);
    lo = hi = 0.0f;
  } else if (r < 2 * (N_ - 1)) {
    int i = r - (N_ - 1);
    zt = xt[xb + N_ + i + 1] - xt[xb + N_ + i]
       - hdt * (xt[xb + 2 * N_ + i] + xt[xb + 2 * N_ + i + 1]);
    lo = hi = 0.0f;
  } else if (r == MEQ_ - 2) { zt = xt[xb + 0];  lo = hi = ic[b * 2 + 0]; }
  else if (r == MEQ_ - 1)   { zt = xt[xb + N_]; lo = hi = ic[b * 2 + 1]; }
  else {
    int j = r - MEQ_;
    zt = xt[xb + j];
    if (j < N_) { lo = -5.0f; hi = 5.0f; } else { lo = -10.0f; hi = 10.0f; }
  }
  float zold = z[zb + r], yold = yv[zb + r];
  float zhat = ALP_ * zt + (1.0f - ALP_) * zold;
  float zn = fminf(fmaxf(zhat + yold / RHO_, lo), hi);
  yv[zb + r] = yold + RHO_ * (zhat - zn);
  z[zb + r] = zn;
}

// ---------------- final heads: mean/std/values ----------------
__global__ __launch_bounds__(128)
void heads_kernel(const float* __restrict__ yb, const float* __restrict__ sb,
                  const float* __restrict__ wb,
                  const float* __restrict__ Wm, const float* __restrict__ bm,
                  const float* __restrict__ Ws, const float* __restrict__ bs,
                  const float* __restrict__ Wv, const float* __restrict__ bv,
                  float* __restrict__ out) {
  __shared__ float red[3][F_];
  int b = blockIdx.x, t = threadIdx.x;
  size_t o = (size_t)b * F_ + t;
  red[0][t] = yb[o] * Wm[t];
  red[1][t] = sb[o] * Ws[t];
  red[2][t] = wb[o] * Wv[t];
  __syncthreads();
  for (int s = 64; s > 0; s >>= 1) {
    if (t < s) {
      red[0][t] += red[0][t + s];
      red[1][t] += red[1][t + s];
      red[2][t] += red[2][t + s];
    }
    __syncthreads();
  }
  if (t == 0) {
    out[b]          = 2.0f * tanhf(red[0][0] + bm[0]);
    out[B_ + b]     = softplusf(red[1][0] + bs[0]);
    out[2 * B_ + b] = red[2][0] + bv[0];
  }
}

// =============================== host side ===============================
extern "C" void kernel_launch(void* const* d_in, const int* in_sizes, int n_in,
                              void* d_out, int out_size, void* d_ws, size_t ws_size,
                              hipStream_t stream) {
  (void)in_sizes; (void)n_in; (void)out_size; (void)ws_size;
  const float* x0 = (const float*)d_in[0];
  const float* W1 = (const float*)d_in[1];  const float* b1 = (const float*)d_in[2];
  const float* W2 = (const float*)d_in[3];  const float* b2 = (const float*)d_in[4];
  const float* Wt = (const float*)d_in[5];  const float* bt = (const float*)d_in[6];
  const float* W4 = (const float*)d_in[7];  const float* b4 = (const float*)d_in[8];
  const float* W5 = (const float*)d_in[9];  const float* b5 = (const float*)d_in[10];
  const float* W6 = (const float*)d_in[11]; const float* b6 = (const float*)d_in[12];
  const float* W7 = (const float*)d_in[13]; const float* b7 = (const float*)d_in[14];
  const float* Wm = (const float*)d_in[15]; const float* bm = (const float*)d_in[16];
  const float* Ws = (const float*)d_in[17]; const float* bs = (const float*)d_in[18];
  const float* Wv = (const float*)d_in[19]; const float* bv = (const float*)d_in[20];
  float* out = (float*)d_out;

  float* ws = (float*)d_ws;
  size_t off = 0;
  auto alloc = [&](size_t n) {
    float* p = ws + off;
    off += (n + 63) & ~(size_t)63;
    return p;
  };
  float* gj   = alloc((size_t)NPAD * GJLD);   // [K | I] -> [I | Kinv]
  float* Amat = alloc((size_t)MROWS * NPAD);
  float* W4p  = alloc((size_t)KACC * F_);
  float* W5p  = alloc((size_t)KACC * F_);
  float* tgt  = alloc(B_);
  float* h1   = alloc((size_t)B_ * F_);
  float* h2   = alloc((size_t)B_ * F_);
  float* w1b  = alloc((size_t)B_ * F_);
  float* w2b  = alloc((size_t)B_ * F_);
  float* yb   = alloc((size_t)B_ * F_);
  float* sb   = alloc((size_t)B_ * F_);
  float* X    = alloc((size_t)B_ * NPAD);
  float* XT   = alloc((size_t)B_ * NPAD);
  float* RHS  = alloc((size_t)B_ * NPAD);
  float* Z    = alloc((size_t)B_ * MPAD);
  float* Y    = alloc((size_t)B_ * MPAD);

  auto zfill = [&](float* p, size_t n) {
    int blocks = (int)((n + 255) / 256); if (blocks > 4096) blocks = 4096;
    zero_kernel<<<blocks, 256, 0, stream>>>(p, n);
  };

  // --- one-time (per launch) QP factorization ---
  zfill(Amat, (size_t)MROWS * NPAD);
  build_A_kernel<<<(MROWS + 255) / 256, 256, 0, stream>>>(Amat);
  build_K_kernel<<<(NPAD * NPAD + 255) / 256, 256, 0, stream>>>(Amat, gj);
  gauss_jordan_kernel<<<1, 1024, 0, stream>>>(gj);
  pad_w_kernel<<<(KACC * F_ + 255) / 256, 256, 0, stream>>>(W4, W5, W4p, W5p);

  // --- head MLP ---
  head1_kernel<<<(B_ * F_ + 255) / 256, 256, 0, stream>>>(x0, W1, b1, h1);
  gemm_feat_kernel<<<dim3(F_ / 32, B_ / 32), 32, 0, stream>>>(h1, F_, 0, W2, b2, h2, F_, 1);
  target_kernel<<<B_, 128, 0, stream>>>(h2, Wt, bt, tgt);
  gemm_feat_kernel<<<dim3(F_ / 32, B_ / 32), 32, 0, stream>>>(h2, F_, 0, W6, b6, w1b, F_, 1);
  gemm_feat_kernel<<<dim3(F_ / 32, B_ / 32), 32, 0, stream>>>(w1b, F_, 0, W7, b7, w2b, F_, 1);

  // --- ADMM state ---
  zfill(X, (size_t)B_ * NPAD);
  zfill(Z, (size_t)B_ * MPAD);
  zfill(Y, (size_t)B_ * MPAD);

  for (int it = 0; it < 20; ++it) {
    admm_rhs_kernel<<<dim3(3, B_), 128, 0, stream>>>(X, Z, Y, tgt, RHS);
    admm_gemm_kernel<<<dim3(NPAD / 32, B_ / 32), 32, 0, stream>>>(RHS, gj, XT, X);
    admm_update_kernel<<<dim3(4, B_), 128, 0, stream>>>(XT, x0, Z, Y);
  }

  // --- tail: acc = X[:,202:303], read via padded-K GEMM starting at col 200 ---
  gemm_feat_kernel<<<dim3(F_ / 32, B_ / 32), 32, 0, stream>>>(X, NPAD, 200, W4p, b4, yb, KACC, 1);
  gemm_feat_kernel<<<dim3(F_ / 32, B_ / 32), 32, 0, stream>>>(X, NPAD, 200, W5p, b5, sb, KACC, 1);
  heads_kernel<<<B_, 128, 0, stream>>>(yb, sb, w2b, Wm, bm, Ws, bs, Wv, bv, out);
}